// ChemicalComputingSubstrate_4681514352755
// MI455X (gfx1250) — compile-verified
//
#include <hip/hip_runtime.h>
#include <math.h>
#include <stdint.h>

#define S_DIM 256
#define R_DIM 2048
#define H_DIM 4096
#define B_DIM 16
#define L_DIM 1024
#define DT_F  0.01f
#define NUM_STEPS 10

typedef __attribute__((ext_vector_type(2))) float v2f;
typedef __attribute__((ext_vector_type(4))) float v4f;
typedef __attribute__((ext_vector_type(8))) float v8f;

// ---------------------------------------------------------------------------
// 1a) partial mean over sequence dim, split into 8 l-chunks for occupancy:
//     part[b][c][h] = sum_{l in chunk c} x[b,l,h]
// grid: (H/1024, B, 8), block 256, b128 non-temporal loads (256MB one-shot
// stream > 192MB L2 -> NT avoids evicting the GEMM working set).
// ---------------------------------------------------------------------------
__global__ void seq_mean_part_kernel(const float* __restrict__ x, float* __restrict__ part) {
    int h4 = blockIdx.x * 256 + threadIdx.x;     // float4 index 0..1023
    int b  = blockIdx.y;
    int c  = blockIdx.z;                         // l-chunk 0..7
    const v4f* p = (const v4f*)(x + (size_t)b * L_DIM * H_DIM) + h4;
    p += (size_t)(c * (L_DIM / 8)) * (H_DIM / 4);
    v4f acc = {0.f, 0.f, 0.f, 0.f};
#pragma unroll 4
    for (int l = 0; l < L_DIM / 8; ++l) {
        acc += __builtin_nontemporal_load(p);
        p += H_DIM / 4;
    }
    ((v4f*)part)[((size_t)b * 8 + c) * (H_DIM / 4) + h4] = acc;
}

// 1b) m[b,h] = (1/L) * sum_c part[b][c][h]   (fixed order -> deterministic)
__global__ void seq_mean_reduce_kernel(const float* __restrict__ part, float* __restrict__ m) {
    int h4 = blockIdx.x * 256 + threadIdx.x;
    int b  = blockIdx.y;
    const v4f* p = (const v4f*)part + (size_t)b * 8 * (H_DIM / 4) + h4;
    v4f acc = {0.f, 0.f, 0.f, 0.f};
#pragma unroll
    for (int c = 0; c < 8; ++c) acc += p[(size_t)c * (H_DIM / 4)];
    ((v4f*)m)[(size_t)b * (H_DIM / 4) + h4] = acc * (1.0f / (float)L_DIM);
}

// ---------------------------------------------------------------------------
// 2) conc0[s] = concentrations[s] + (1/B) * sum_b softplus(m[b,:]@w[:,s] + bias[s])
// grid: S blocks, 256 threads. Deterministic fixed-order LDS tree reductions.
// ---------------------------------------------------------------------------
__global__ void n2c_kernel(const float* __restrict__ m, const float* __restrict__ w,
                           const float* __restrict__ bias, const float* __restrict__ conc_in,
                           float* __restrict__ conc0) {
    int s = blockIdx.x;
    int t = threadIdx.x;
    __shared__ float red[256];
    float partial[B_DIM];
#pragma unroll
    for (int b = 0; b < B_DIM; ++b) partial[b] = 0.f;
    for (int i = 0; i < H_DIM / 256; ++i) {
        int h = t + i * 256;
        float wv = w[(size_t)h * S_DIM + s];
#pragma unroll
        for (int b = 0; b < B_DIM; ++b) partial[b] += m[b * H_DIM + h] * wv;
    }
    float tot = 0.f;
    for (int b = 0; b < B_DIM; ++b) {
        red[t] = partial[b];
        __syncthreads();
        for (int off = 128; off > 0; off >>= 1) {
            if (t < off) red[t] += red[t + off];
            __syncthreads();
        }
        if (t == 0) {
            float x = red[0] + bias[s];
            tot += fmaxf(x, 0.f) + log1pf(expf(-fabsf(x)));  // stable softplus
        }
        __syncthreads();
    }
    if (t == 0) conc0[s] = conc_in[s] + tot * (1.0f / (float)B_DIM);
}

// ---------------------------------------------------------------------------
// 3) f0[j] = prod_s conc0[s]^max(stoich[j,s],0). One wave per reaction row.
// ---------------------------------------------------------------------------
__global__ void f0_kernel(const float* __restrict__ conc0, const float* __restrict__ stoich,
                          float* __restrict__ f) {
    int wave = threadIdx.x >> 5;
    int lane = threadIdx.x & 31;
    int j = blockIdx.x * 8 + wave;
    const float* row = stoich + (size_t)j * S_DIM;
    float prod = 1.f;
#pragma unroll
    for (int i = 0; i < S_DIM / 32; ++i) {
        int s = i * 32 + lane;
        float p = fmaxf(row[s], 0.f);
        prod *= powf(conc0[s], p);
    }
    for (int off = 16; off > 0; off >>= 1)
        prod *= __shfl_xor(prod, off, 32);
    if (lane == 0) f[j] = prod;
}

// ---------------------------------------------------------------------------
// 4) g[s] = sum_j f[j] * stoich[j,s]  (rank-1 factorization of rates@stoich)
// ---------------------------------------------------------------------------
__global__ void g_kernel(const float* __restrict__ f, const float* __restrict__ stoich,
                         float* __restrict__ g) {
    int s = blockIdx.x;
    int t = threadIdx.x;
    __shared__ float red[256];
    float acc = 0.f;
#pragma unroll
    for (int i = 0; i < R_DIM / 256; ++i) {
        int j = t + i * 256;
        acc += f[j] * stoich[(size_t)j * S_DIM + s];
    }
    red[t] = acc;
    __syncthreads();
    for (int off = 128; off > 0; off >>= 1) {
        if (t < off) red[t] += red[t + off];
        __syncthreads();
    }
    if (t == 0) g[s] = red[0];
}

// ---------------------------------------------------------------------------
// 5) fused update + next-step row product. grid: R blocks, 256 threads.
// ---------------------------------------------------------------------------
__global__ void uf_kernel(const float* __restrict__ conc0, float* __restrict__ conc,
                          const float* __restrict__ g, const float* __restrict__ k,
                          const float* __restrict__ stoich, float* __restrict__ f, int first) {
    int i = blockIdx.x;
    int s = threadIdx.x;
    __shared__ float red[256];
    float cin = first ? conc0[s] : conc[(size_t)i * S_DIM + s];
    float v = cin + DT_F * k[i] * g[s];
    v = fminf(fmaxf(v, 0.f), 10.f);
    conc[(size_t)i * S_DIM + s] = v;
    float p = fmaxf(stoich[(size_t)i * S_DIM + s], 0.f);
    red[s] = powf(v, p);
    __syncthreads();
    for (int off = 128; off > 0; off >>= 1) {
        if (s < off) red[s] *= red[s + off];
        __syncthreads();
    }
    if (s == 0) f[i] = red[0];
}

// ---------------------------------------------------------------------------
// 6) f32 WMMA GEMM with double-buffered async global->LDS tile copies.
//    C = act(A[M,K] @ B[K,N] + bias[N]).
// block = 256 threads = 8 waves; block tile 128x128, K-tile 16.
// Wave w computes rows [w*16, w*16+16) x 128 cols (8 v8f accumulators).
// A tile stored chunk-major [4 kchunks][128 rows][4 floats]:
//   - async b128 copies are contiguous 16B chunks (engine-compatible)
//   - A-frag read = lane-stride-4 -> conflict-free, (k,k+1) pair = one ds_load_b64
// Tiles fetched with GLOBAL_LOAD_ASYNC_TO_LDS_B128 (ASYNCcnt), next tile
// streams from L2 while the current one feeds V_WMMA_F32_16X16X4_F32.
// ---------------------------------------------------------------------------
template <bool GELU>
__global__ void gemm_wmma_kernel(const float* __restrict__ A, const float* __restrict__ Bm,
                                 const float* __restrict__ bias, float* __restrict__ C,
                                 int M, int N, int K) {
    constexpr int BM = 128, BN = 128, BK = 16;
    __shared__ float sA[2][4 * 128 * 4];   // [chunk][row][4] chunk-major
    __shared__ float sB[2][BK * BN];       // [k][n] row-major

    const int tid   = threadIdx.x;
    const int wave  = tid >> 5;
    const int lane  = tid & 31;
    const int m0    = blockIdx.y * BM;
    const int n0    = blockIdx.x * BN;
    const int lan16 = lane & 15;
    const int kpair = (lane >> 4) * 2;   // lanes 0-15: K 0,1 ; lanes 16-31: K 2,3
    const int mrow  = wave * 16;

    // async copy mappings (per thread: 2 A chunks + 2 B chunks = 4 x b128)
    const int ar  = tid >> 1;          // A tile row 0..127
    const int ac0 = (tid & 1) * 2;     // first A chunk (0 or 2)
    const int br  = tid >> 4;          // B tile row 0..15
    const int bn0 = (tid & 15) * 8;    // B tile col 0..120

    const uint32_t a_dst[2] = {
        (uint32_t)(uintptr_t)&sA[0][ac0 * 512 + ar * 4],
        (uint32_t)(uintptr_t)&sA[1][ac0 * 512 + ar * 4]};
    const uint32_t b_dst[2] = {
        (uint32_t)(uintptr_t)&sB[0][br * BN + bn0],
        (uint32_t)(uintptr_t)&sB[1][br * BN + bn0]};

    v8f zero = {0.f, 0.f, 0.f, 0.f, 0.f, 0.f, 0.f, 0.f};
    v8f acc[8] = {zero, zero, zero, zero, zero, zero, zero, zero};

    auto issue_tile = [&](int k0, int buf) {
        const float* asrc = A + (size_t)(m0 + ar) * K + k0 + ac0 * 4;
        const float* bsrc = Bm + (size_t)(k0 + br) * N + n0 + bn0;
        uint32_t ad = a_dst[buf];
        uint32_t bd = b_dst[buf];
        asm volatile("global_load_async_to_lds_b128 %0, %1, off"
                     :: "v"(ad), "v"(asrc) : "memory");
        asm volatile("global_load_async_to_lds_b128 %0, %1, off"
                     :: "v"((uint32_t)(ad + 2048u)), "v"(asrc + 4) : "memory");
        asm volatile("global_load_async_to_lds_b128 %0, %1, off"
                     :: "v"(bd), "v"(bsrc) : "memory");
        asm volatile("global_load_async_to_lds_b128 %0, %1, off"
                     :: "v"((uint32_t)(bd + 16u)), "v"(bsrc + 4) : "memory");
    };

    issue_tile(0, 0);
    const int T = K / BK;
    for (int kt = 0; kt < T; ++kt) {
        __syncthreads();  // all waves done reading buf[(kt+1)&1] (iter kt-1)
        if (kt + 1 < T) {
            issue_tile((kt + 1) * BK, (kt + 1) & 1);
            // in-order async completion: <=4 outstanding => tile kt resident
            asm volatile("s_wait_asynccnt 0x4" ::: "memory");
        } else {
            asm volatile("s_wait_asynccnt 0x0" ::: "memory");
        }
        __syncthreads();  // tile kt visible to all waves

        const float* sAb = sA[kt & 1];
        const float* sBb = sB[kt & 1];
#pragma unroll
        for (int kk = 0; kk < 4; ++kk) {
            const int kb = kk * 4 + kpair;
            v2f a = *(const v2f*)&sAb[kk * 512 + (mrow + lan16) * 4 + kpair];
#pragma unroll
            for (int nt = 0; nt < 8; ++nt) {
                v2f b;
                b.x = sBb[kb * BN + nt * 16 + lan16];
                b.y = sBb[(kb + 1) * BN + nt * 16 + lan16];
                acc[nt] = __builtin_amdgcn_wmma_f32_16x16x4_f32(
                    false, a, false, b, (short)0, acc[nt], false, false);
            }
        }
    }

    // Epilogue: bias (+ exact GELU); D layout: vgpr v -> row v + 8*(lane>>4)
    const int rowhalf = (lane >> 4) * 8;
#pragma unroll
    for (int nt = 0; nt < 8; ++nt) {
        int col = n0 + nt * 16 + lan16;
        float bv = bias[col];
#pragma unroll
        for (int v = 0; v < 8; ++v) {
            int row = m0 + mrow + rowhalf + v;
            float val = acc[nt][v] + bv;
            if (GELU) val = 0.5f * val * (1.0f + erff(val * 0.70710678118654752f));
            C[(size_t)row * N + col] = val;
        }
    }
}

// ---------------------------------------------------------------------------
extern "C" void kernel_launch(void* const* d_in, const int* in_sizes, int n_in,
                              void* d_out, int out_size, void* d_ws, size_t ws_size,
                              hipStream_t stream) {
    const float* neural_input   = (const float*)d_in[0];
    const float* concentrations = (const float*)d_in[1];
    const float* stoich         = (const float*)d_in[2];
    const float* rate_k         = (const float*)d_in[3];
    const float* w_n2c          = (const float*)d_in[4];
    const float* b_n2c          = (const float*)d_in[5];
    const float* w_c2n1         = (const float*)d_in[6];
    const float* b_c2n1         = (const float*)d_in[7];
    const float* w_c2n2         = (const float*)d_in[8];
    const float* b_c2n2         = (const float*)d_in[9];
    float* out = (float*)d_out;

    float* ws    = (float*)d_ws;
    float* m     = ws;                          // B*H   = 65536
    float* conc0 = m + B_DIM * H_DIM;           // S     = 256
    float* f     = conc0 + S_DIM;               // R     = 2048
    float* g     = f + R_DIM;                   // S     = 256
    float* conc  = g + S_DIM;                   // R*S   = 524288
    float* h     = conc + R_DIM * S_DIM;        // R*H   = 8388608
    float* part  = h;  // 8*B*H = 524288 floats; aliases h (h unused until GEMM1)

    seq_mean_part_kernel<<<dim3(H_DIM / 1024, B_DIM, 8), 256, 0, stream>>>(neural_input, part);
    seq_mean_reduce_kernel<<<dim3(H_DIM / 1024, B_DIM), 256, 0, stream>>>(part, m);
    n2c_kernel<<<S_DIM, 256, 0, stream>>>(m, w_n2c, b_n2c, concentrations, conc0);
    f0_kernel<<<R_DIM / 8, 256, 0, stream>>>(conc0, stoich, f);
    for (int t = 0; t < NUM_STEPS; ++t) {
        g_kernel<<<S_DIM, 256, 0, stream>>>(f, stoich, g);
        uf_kernel<<<R_DIM, 256, 0, stream>>>(conc0, conc, g, rate_k, stoich, f, t == 0);
    }
    // h = gelu(conc @ w_c2n1 + b_c2n1)   [2048,256]x[256,4096]
    gemm_wmma_kernel<true><<<dim3(H_DIM / 128, R_DIM / 128), 256, 0, stream>>>(
        conc, w_c2n1, b_c2n1, h, R_DIM, H_DIM, S_DIM);
    // out = h @ w_c2n2 + b_c2n2          [2048,4096]x[4096,4096]
    gemm_wmma_kernel<false><<<dim3(H_DIM / 128, R_DIM / 128), 256, 0, stream>>>(
        h, w_c2n2, b_c2n2, out, R_DIM, H_DIM, H_DIM);
    // second output: final concentrations [R,S]
    hipMemcpyAsync(out + (size_t)R_DIM * H_DIM, conc,
                   sizeof(float) * (size_t)R_DIM * S_DIM,
                   hipMemcpyDeviceToDevice, stream);
}